// DRMM_6090263625992
// MI455X (gfx1250) — compile-verified
//
#include <hip/hip_runtime.h>
#include <math.h>

// Problem constants (match the reference).
constexpr int kB  = 32;
constexpr int kQ  = 16;
constexpr int kL  = 4096;
constexpr int kE  = 300;     // 75 * 4 -> exact fit for WMMA f32 K=4
constexpr int kNB = 30;
constexpr int kSeg   = 16;           // L-segments per batch row (occupancy)
constexpr int kTiles = kL / 16;      // 256 doc tiles of 16 tokens

typedef __attribute__((ext_vector_type(2))) float v2f;
typedef __attribute__((ext_vector_type(8))) float v8f;

// Bin one 16x16 accumulator tile into the LDS histogram.
// C/D layout: VGPR v -> (q = v + 8*half, doc col = lane&15).
__device__ __forceinline__ void bin_tile(
    const v8f& c, float ndp, const float* nqinv, int half,
    unsigned int* hist)
{
  // combine the two K sub-phases of |doc row|^2, then fast rsqrt.
  float nd2 = ndp + __shfl_xor(ndp, 16, 32);
  const float rdinv = __builtin_amdgcn_rsqf(nd2);   // v_rsq_f32 (~1 ulp)
#pragma unroll
  for (int v = 0; v < 8; ++v) {
    // c / max(nq*nd, 1e-8) == c * min(1/(nq*nd), 1e8) (to ~2 ulp; selects a
    // 1/15-wide bin, so approximate rcp/rsq is fine).
    const float inv   = fminf(nqinv[v] * rdinv, 1e8f);
    const float inter = c[v] * inv;
    int bin = (int)floorf((inter + 1.0f) * (kNB * 0.5f));
    bin = bin < 0 ? 0 : (bin > kNB - 1 ? kNB - 1 : bin);
    atomicAdd(&hist[(v + 8 * half) * kNB + bin], 1u);
  }
}

// Core: pairs of 16x16 dot tiles via V_WMMA_F32_16X16X4_F32 with two
// independent accumulator chains; doc norms fused into the load stream.
__device__ __forceinline__ void drmm_tiles(
    const float* __restrict__ qbase,   // query[b] : 16 x 300
    const float* __restrict__ dbase,   // doc[b]   : 4096 x 300
    const float* nq_s,                 // LDS: 16 query norms
    unsigned int* hist,                // LDS: 16*30 counters
    int t0, int t1)                    // tile range, (t1-t0) % 16 == 0
{
  const int tid  = threadIdx.x;
  const int lane = tid & 31;
  const int wave = tid >> 5;           // 8 waves per 256-thread block
  const int half = lane >> 4;          // K sub-phase: 0 -> K%4 in {0,1}, 1 -> {2,3}
  const int l15  = lane & 15;

  float nqinv[8];
#pragma unroll
  for (int v = 0; v < 8; ++v)
    nqinv[v] = __builtin_amdgcn_rcpf(nq_s[v + 8 * half]);  // v_rcp_f32

  // A (query) per-lane row base: lane m holds (K+2*half, K+2*half+1) of row m.
  const float* arow = qbase + (long)l15 * kE + 2 * half;

  for (int t = t0 + wave; t < t1; t += 16) {
    // Two doc tiles 8 apart -> two independent WMMA accumulation chains.
    const float* brow0 = dbase + (long)((t     ) * 16 + l15) * kE + 2 * half;
    const float* brow1 = dbase + (long)((t +  8) * 16 + l15) * kE + 2 * half;

    v8f c0 = {0.f, 0.f, 0.f, 0.f, 0.f, 0.f, 0.f, 0.f};
    v8f c1 = {0.f, 0.f, 0.f, 0.f, 0.f, 0.f, 0.f, 0.f};
    // Per-tile packed norm accumulators: pk_fma operates directly on the
    // loaded register pairs (no cross-accumulator packing movs).
    v2f nd0 = {0.f, 0.f}, nd1 = {0.f, 0.f};

    // One-chunk software pipeline: loads for chunk k+1 are in flight while
    // chunk k feeds the two WMMA chains.
    v2f a  = *(const v2f*)(arow);
    v2f b0 = *(const v2f*)(brow0);
    v2f b1 = *(const v2f*)(brow1);
#pragma unroll 5
    for (int kk = 4; kk < kE; kk += 4) {
      const v2f an  = *(const v2f*)(arow  + kk);
      const v2f bn0 = *(const v2f*)(brow0 + kk);
      const v2f bn1 = *(const v2f*)(brow1 + kk);
      nd0 += b0 * b0;                  // v_pk_fma_f32
      nd1 += b1 * b1;
      c0 = __builtin_amdgcn_wmma_f32_16x16x4_f32(
               false, a, false, b0, (short)0, c0, false, false);
      c1 = __builtin_amdgcn_wmma_f32_16x16x4_f32(
               false, a, false, b1, (short)0, c1, false, false);
      a = an; b0 = bn0; b1 = bn1;
    }
    nd0 += b0 * b0;                    // tail chunk (kk = kE-4)
    nd1 += b1 * b1;
    c0 = __builtin_amdgcn_wmma_f32_16x16x4_f32(
             false, a, false, b0, (short)0, c0, false, false);
    c1 = __builtin_amdgcn_wmma_f32_16x16x4_f32(
             false, a, false, b1, (short)0, c1, false, false);

    bin_tile(c0, nd0.x + nd0.y, nqinv, half, hist);
    bin_tile(c1, nd1.x + nd1.y, nqinv, half, hist);
  }
}

__global__ void drmm_zero_kernel(unsigned int* __restrict__ p, int n) {
  int i = blockIdx.x * blockDim.x + threadIdx.x;
  if (i < n) p[i] = 0u;
}

__global__ __launch_bounds__(256) void drmm_hist_kernel(
    const float* __restrict__ query, const float* __restrict__ doc,
    unsigned int* __restrict__ ghist)
{
  __shared__ unsigned int hist[kQ * kNB];
  __shared__ float nq_s[kQ];
  const int b   = blockIdx.x / kSeg;
  const int seg = blockIdx.x % kSeg;
  const int tid = threadIdx.x;

  for (int i = tid; i < kQ * kNB; i += 256) hist[i] = 0u;
  if (tid < kQ) {                        // query norms (tiny)
    const float* qr = query + ((long)b * kQ + tid) * kE;
    float s = 0.f;
    for (int e = 0; e < kE; ++e) s = fmaf(qr[e], qr[e], s);
    nq_s[tid] = sqrtf(s);
  }
  __syncthreads();

  const int tps = kTiles / kSeg;         // 16 tiles per segment
  drmm_tiles(query + (long)b * kQ * kE, doc + (long)b * kL * kE,
             nq_s, hist, seg * tps, (seg + 1) * tps);
  __syncthreads();

  unsigned int* gh = ghist + (long)b * kQ * kNB;   // merge into global hist
  for (int i = tid; i < kQ * kNB; i += 256)
    if (hist[i]) atomicAdd(&gh[i], hist[i]);
}

__device__ __forceinline__ void drmm_head(
    const float* __restrict__ query, const unsigned int* hq_counts,
    const float* __restrict__ w1, const float* __restrict__ b1,
    const float* __restrict__ w2, const float* __restrict__ b2,
    const float* __restrict__ w3, const float* __restrict__ b3,
    const float* __restrict__ gw, const float* __restrict__ gb,
    float* z_s, float* lg_s, float* __restrict__ out, int b, int tid)
{
  if (tid < kQ) {
    float h[kNB];
    for (int j = 0; j < kNB; ++j) h[j] = log1pf((float)hq_counts[j]);
    float x[5];
    for (int i = 0; i < 5; ++i) {
      float s = b1[i];
      for (int j = 0; j < kNB; ++j) s = fmaf(w1[i * kNB + j], h[j], s);
      x[i] = tanhf(s);
    }
    float y = b2[0];
    for (int i = 0; i < 5; ++i) y = fmaf(w2[i], x[i], y);
    y = tanhf(y);
    z_s[tid] = tanhf(fmaf(w3[0], y, b3[0]));

    const float* qr = query + ((long)b * kQ + tid) * kE;
    float g = gb[0];
    for (int e = 0; e < kE; ++e) g = fmaf(gw[e], qr[e], g);
    lg_s[tid] = g;
  }
  __syncthreads();
  if (tid == 0) {
    float m = lg_s[0];
    for (int q = 1; q < kQ; ++q) m = fmaxf(m, lg_s[q]);
    float s = 0.f, acc = 0.f;
    for (int q = 0; q < kQ; ++q) {
      float e = expf(lg_s[q] - m);
      s += e;
      acc = fmaf(z_s[q], e, acc);
    }
    out[b] = acc / s;
  }
}

__global__ __launch_bounds__(32) void drmm_final_kernel(
    const float* __restrict__ query, const unsigned int* __restrict__ ghist,
    const float* __restrict__ w1, const float* __restrict__ b1,
    const float* __restrict__ w2, const float* __restrict__ b2,
    const float* __restrict__ w3, const float* __restrict__ b3,
    const float* __restrict__ gw, const float* __restrict__ gb,
    float* __restrict__ out)
{
  __shared__ float z_s[kQ], lg_s[kQ];
  const int b = blockIdx.x, tid = threadIdx.x;
  const unsigned int* hq =
      ghist + ((long)b * kQ + (tid < kQ ? tid : 0)) * kNB;
  drmm_head(query, hq, w1, b1, w2, b2, w3, b3, gw, gb, z_s, lg_s, out, b, tid);
}

// Fallback: whole problem in one kernel (grid = B), LDS-only histogram.
__global__ __launch_bounds__(256) void drmm_fused_kernel(
    const float* __restrict__ query, const float* __restrict__ doc,
    const float* __restrict__ w1, const float* __restrict__ b1,
    const float* __restrict__ w2, const float* __restrict__ b2,
    const float* __restrict__ w3, const float* __restrict__ b3,
    const float* __restrict__ gw, const float* __restrict__ gb,
    float* __restrict__ out)
{
  __shared__ unsigned int hist[kQ * kNB];
  __shared__ float nq_s[kQ], z_s[kQ], lg_s[kQ];
  const int b = blockIdx.x, tid = threadIdx.x;

  for (int i = tid; i < kQ * kNB; i += 256) hist[i] = 0u;
  if (tid < kQ) {
    const float* qr = query + ((long)b * kQ + tid) * kE;
    float s = 0.f;
    for (int e = 0; e < kE; ++e) s = fmaf(qr[e], qr[e], s);
    nq_s[tid] = sqrtf(s);
  }
  __syncthreads();

  drmm_tiles(query + (long)b * kQ * kE, doc + (long)b * kL * kE,
             nq_s, hist, 0, kTiles);
  __syncthreads();

  const unsigned int* hq = &hist[(tid < kQ ? tid : 0) * kNB];
  drmm_head(query, hq, w1, b1, w2, b2, w3, b3, gw, gb, z_s, lg_s, out, b, tid);
}

extern "C" void kernel_launch(void* const* d_in, const int* in_sizes, int n_in,
                              void* d_out, int out_size, void* d_ws, size_t ws_size,
                              hipStream_t stream) {
  const float* query = (const float*)d_in[0];
  const float* doc   = (const float*)d_in[1];
  const float* w1    = (const float*)d_in[2];
  const float* b1    = (const float*)d_in[3];
  const float* w2    = (const float*)d_in[4];
  const float* b2    = (const float*)d_in[5];
  const float* w3    = (const float*)d_in[6];
  const float* b3    = (const float*)d_in[7];
  const float* gw    = (const float*)d_in[8];
  const float* gb    = (const float*)d_in[9];
  float* out = (float*)d_out;

  const size_t hist_bytes = (size_t)kB * kQ * kNB * sizeof(unsigned int);
  if (ws_size >= hist_bytes) {
    unsigned int* ghist = (unsigned int*)d_ws;
    const int n = kB * kQ * kNB;
    drmm_zero_kernel<<<(n + 255) / 256, 256, 0, stream>>>(ghist, n);
    drmm_hist_kernel<<<kB * kSeg, 256, 0, stream>>>(query, doc, ghist);
    drmm_final_kernel<<<kB, 32, 0, stream>>>(query, ghist, w1, b1, w2, b2,
                                             w3, b3, gw, gb, out);
  } else {
    drmm_fused_kernel<<<kB, 256, 0, stream>>>(query, doc, w1, b1, w2, b2,
                                              w3, b3, gw, gb, out);
  }
}